// TokenAwareFA_44598940402135
// MI455X (gfx1250) — compile-verified
//
#include <hip/hip_runtime.h>
#include <hip/hip_bf16.h>
#include <stdint.h>

// Memory-bound bilinear-resample + weighted group average.
// features[(b*N + n)*1024 + (y*32 + x)]*1024 + d  ->  out[b*1024 + p]*1024 + d
// One block (256 thr) per output pixel; thread t owns channels 4t..4t+3.
// Group-0 (identity) term is staged via the CDNA5 async global->LDS DMA path
// (ASYNCcnt, th:NT) and overlapped with the 12 branchless bilinear gathers.

typedef float v4f __attribute__((ext_vector_type(4)));

template <int NS>
__global__ __launch_bounds__(256) void TokenAwareFA_kernel(
    const float* __restrict__ features,
    const float* __restrict__ inv_affine,   // (N, 2, 3)
    const float* __restrict__ rot_w,        // (N,)
    float* __restrict__ out,
    int Nrt)
{
    constexpr int H = 32, W = 32, D = 1024;
    constexpr int P = H * W;
    const int N = (NS > 0) ? NS : Nrt;

    __shared__ float lds0[256 * 4];   // 4 KB: group-0 slice for this pixel

    const int bid = blockIdx.x;
    const int b   = bid >> 10;        // / P
    const int p   = bid & (P - 1);
    const int h   = p >> 5;
    const int w   = p & 31;
    const int tid = threadIdx.x;
    const int d   = tid << 2;         // 4 channels per thread

    // ---- kick off async DMA of the identity (group 0) term into LDS ----
    const float* src0 = features + ((size_t)(b * N) * P + p) * (size_t)D + d;
    uint32_t ldsaddr = (uint32_t)(uintptr_t)(&lds0[tid * 4]);
    asm volatile("global_load_async_to_lds_b128 %0, %1, off th:TH_LOAD_NT"
                 :: "v"(ldsaddr), "v"(src0)
                 : "memory");

    // ---- branchless bilinear gathers for rotated groups (overlap with DMA) ----
    const float step = 2.0f / 31.0f;                 // matches jnp.linspace(-1,1,32)
    const float fx = -1.0f + step * (float)w;
    const float fy = -1.0f + step * (float)h;

    v4f acc = (v4f)0.0f;

    #pragma unroll
    for (int n = 1; n < N; ++n) {
        const float t00 = inv_affine[n * 6 + 0];
        const float t01 = inv_affine[n * 6 + 1];
        const float t02 = inv_affine[n * 6 + 2];
        const float t10 = inv_affine[n * 6 + 3];
        const float t11 = inv_affine[n * 6 + 4];
        const float t12 = inv_affine[n * 6 + 5];
        const float wn  = rot_w[n];

        const float gx = (t00 * fx + t01 * fy + t02 + 1.0f) * 0.5f * 31.0f;
        const float gy = (t10 * fx + t11 * fy + t12 + 1.0f) * 0.5f * 31.0f;
        const float x0 = floorf(gx), y0 = floorf(gy);
        const float wx = gx - x0,    wy = gy - y0;

        float cw[4] = { (1.0f - wx) * (1.0f - wy) * wn,
                        wx          * (1.0f - wy) * wn,
                        (1.0f - wx) * wy          * wn,
                        wx          * wy          * wn };
        const float tx[4] = { x0, x0 + 1.0f, x0,        x0 + 1.0f };
        const float ty[4] = { y0, y0,        y0 + 1.0f, y0 + 1.0f };

        const float* srcn = features + ((size_t)(b * N + n) * P) * (size_t)D + d;

        #pragma unroll
        for (int t = 0; t < 4; ++t) {
            // Reference semantics: clip index, zero the contribution if OOB.
            const bool valid = (tx[t] >= 0.0f) & (tx[t] <= 31.0f) &
                               (ty[t] >= 0.0f) & (ty[t] <= 31.0f);
            const float c = valid ? cw[t] : 0.0f;
            int xi = (int)fminf(fmaxf(tx[t], 0.0f), 31.0f);
            int yi = (int)fminf(fmaxf(ty[t], 0.0f), 31.0f);
            const v4f v = *(const v4f*)(srcn + (size_t)(yi * 32 + xi) * (size_t)D);
            acc += c * v;
        }
    }

    // ---- fold in the async-staged identity term ----
    asm volatile("s_wait_asynccnt 0x0" ::: "memory");
    const float w0 = rot_w[0];
    const v4f v0 = *(const v4f*)(&lds0[tid * 4]);   // ds_load_b128
    acc += w0 * v0;

    // Output is write-once: nontemporal b128 store keeps L2 for the gather taps.
    v4f* o = (v4f*)(out + ((size_t)b * P + p) * (size_t)D + d);
    __builtin_nontemporal_store(acc, o);
}

extern "C" void kernel_launch(void* const* d_in, const int* in_sizes, int n_in,
                              void* d_out, int out_size, void* d_ws, size_t ws_size,
                              hipStream_t stream) {
    const float* features   = (const float*)d_in[0];
    const float* inv_affine = (const float*)d_in[1];
    const float* rot_w      = (const float*)d_in[2];
    // d_in[3] = batch_size scalar (device); shapes derived host-side instead.
    float* out = (float*)d_out;

    const int N = in_sizes[1] / 6;                       // (N,2,3) -> N
    const int B = in_sizes[0] / (N * 1024 * 1024);       // B*N*1024*1024 elems
    const int blocks = B * 1024;                         // one block per pixel

    if (N == 4) {
        TokenAwareFA_kernel<4><<<blocks, 256, 0, stream>>>(features, inv_affine,
                                                           rot_w, out, N);
    } else {
        TokenAwareFA_kernel<0><<<blocks, 256, 0, stream>>>(features, inv_affine,
                                                           rot_w, out, N);
    }
}